// QuantumPatchEncoder_65481071400241
// MI455X (gfx1250) — compile-verified
//
#include <hip/hip_runtime.h>

typedef float v2f __attribute__((ext_vector_type(2)));
typedef float v8f __attribute__((ext_vector_type(8)));

#define N_WIRES 4
#define NUM_PARAMS 8

// ---------------------------------------------------------------------------
// Kernel 1: build the fixed 16x16 variational-layer matrix U from weights.
// Each of 16 lanes pushes one basis column e_l through the gate sequence.
// U[k][l] = <k| (gate sequence) |l>.  Wire w <-> bit (3-w) of the state index.
// ---------------------------------------------------------------------------
__global__ void build_unitary_kernel(const float* __restrict__ w,
                                     float* __restrict__ U) {
    const int l = threadIdx.x;
    if (l >= 16) return;
    float a[16];
#pragma unroll
    for (int n = 0; n < 16; ++n) a[n] = (n == l) ? 1.0f : 0.0f;

#pragma unroll
    for (int i = 0; i < NUM_PARAMS; ++i) {
        const int wr = i & 3;
        const float half = 0.5f * w[i];
        const float c = __cosf(half);
        const float s = __sinf(half);
        const int m = 8 >> wr;  // bit mask for wire wr
        // RY(theta) on wire wr
#pragma unroll
        for (int n = 0; n < 16; ++n) {
            if (!(n & m)) {
                const float a0 = a[n];
                const float a1 = a[n | m];
                a[n]     = c * a0 - s * a1;
                a[n | m] = s * a0 + c * a1;
            }
        }
        // CNOT(control = wr, target = (i+1)%4): swap amplitudes where
        // control bit is set (target-flip permutation, its own inverse).
        const int cm = 8 >> wr;
        const int tm = 8 >> ((i + 1) & 3);
#pragma unroll
        for (int n = 0; n < 16; ++n) {
            if ((n & cm) && !(n & tm)) {
                const float t = a[n];
                a[n] = a[n | tm];
                a[n | tm] = t;
            }
        }
    }
#pragma unroll
    for (int k = 0; k < 16; ++k) U[k * 16 + l] = a[k];
}

// ---------------------------------------------------------------------------
// Kernel 2: persistent waves, one 16-row batch tile per loop iteration.
// GEMM orientation: D[k_amp, row] = sum_j U[k_amp, j] * s_enc[row, j]
//   A (16x16, chunked 16x4)  = U        -> lane%16 = M=k_amp, K = 4c + 2h + v
//   B (16x16, chunked 4x16)  = s_enc^T  -> lane%16 = N=row,   K = 4c + 2h + v
//   C/D: VGPR v, lanes 0-15: amp = v, row = lane; lanes 16-31: amp = v+8.
//
// Key layout trick: the K index this lane supplies is n = 4c + 2h + v, so
// bit1(n) == h for the whole lane.  With amp[n] = q01[n>>2]*g2[bit1]*g3[bit0]
// the only lane-dependent choice is the single g2[h] factor -> one cndmask,
// no dynamic register-array indexing.
// ---------------------------------------------------------------------------
__global__ void __launch_bounds__(256)
qenc_wmma_kernel(const float* __restrict__ in, const float* __restrict__ U,
                 float* __restrict__ out, int tiles) {
    const int lane   = threadIdx.x & 31;
    const int waveId = (blockIdx.x * blockDim.x + threadIdx.x) >> 5;
    const int nWaves = (gridDim.x * blockDim.x) >> 5;
    const int m = lane & 15;           // row-within-tile / M index
    const bool hi = (lane >= 16);      // wave half -> bit1 of K index

    // Load A = U into WMMA A-layout registers once per wave (reused all tiles).
    v2f A[4];
#pragma unroll
    for (int c = 0; c < 4; ++c) {
        const int col = 4 * c + (hi ? 2 : 0);
        A[c].x = U[m * 16 + col];
        A[c].y = U[m * 16 + col + 1];
    }

    for (int tile = waveId; tile < tiles; tile += nWaves) {
        const int row = tile * 16 + m;
        const float4 x = ((const float4*)in)[row];

        // Per-wire encoding factors f_w = (cos(x_w/2), sin(x_w/2)).
        const float g0c = __cosf(0.5f * x.x), g0s = __sinf(0.5f * x.x);
        const float g1c = __cosf(0.5f * x.y), g1s = __sinf(0.5f * x.y);
        const float g2c = __cosf(0.5f * x.z), g2s = __sinf(0.5f * x.z);
        const float g3c = __cosf(0.5f * x.w), g3s = __sinf(0.5f * x.w);

        // q01[c] = g0[bit3]*g1[bit2] for n>>2 == c  (compile-time indexed).
        float q01[4];
        q01[0] = g0c * g1c;
        q01[1] = g0c * g1s;
        q01[2] = g0s * g1c;
        q01[3] = g0s * g1s;

        // Lane-half factor: bit1(n) == hi for every K element of this lane.
        const float g2h = hi ? g2s : g2c;   // single cndmask
        const float e0 = g2h * g3c;         // bit0 = 0 elements
        const float e1 = g2h * g3s;         // bit0 = 1 elements

        // final = U * s_enc via 4 chained f32 WMMAs (K = 16).
        v8f acc = {};
#pragma unroll
        for (int c = 0; c < 4; ++c) {
            v2f b;
            b.x = q01[c] * e0;
            b.y = q01[c] * e1;
            acc = __builtin_amdgcn_wmma_f32_16x16x4_f32(
                /*neg_a=*/false, A[c], /*neg_b=*/false, b,
                /*c_mod=*/(short)0, acc, /*reuse_a=*/false, /*reuse_b=*/false);
        }

        // Probabilities for amps n = v + 8*hi held in this lane.
        float p[8];
#pragma unroll
        for (int v = 0; v < 8; ++v) p[v] = acc[v] * acc[v];

        // <Z_w>: sign = bit (3-w) of n.  Bits 2..0 come from v (compile-time
        // signs, shared partial sums); bit 3 = half, handled at the combine.
        const float s01 = p[0] + p[1], s23 = p[2] + p[3];
        const float s45 = p[4] + p[5], s67 = p[6] + p[7];
        const float a03 = s01 + s23,   a47 = s45 + s67;
        const float t0 = a03 + a47;
        const float t1 = a03 - a47;
        const float t2 = (s01 + s45) - (s23 + s67);
        const float t3 = (p[0] + p[2] + p[4] + p[6]) - (p[1] + p[3] + p[5] + p[7]);

        // Combine amp halves (lane r <-> lane r+16).
        const float u0 = __shfl_xor(t0, 16, 32);
        const float u1 = __shfl_xor(t1, 16, 32);
        const float u2 = __shfl_xor(t2, 16, 32);
        const float u3 = __shfl_xor(t3, 16, 32);

        if (lane < 16) {
            float4 o;
            o.x = t0 - u0;   // wire 0: upper-half amps carry sign -1
            o.y = t1 + u1;
            o.z = t2 + u2;
            o.w = t3 + u3;
            ((float4*)out)[row] = o;
        }
        // predicate closes here; EXEC is all-ones again before next WMMA
    }
}

extern "C" void kernel_launch(void* const* d_in, const int* in_sizes, int n_in,
                              void* d_out, int out_size, void* d_ws, size_t ws_size,
                              hipStream_t stream) {
    const float* inputs  = (const float*)d_in[0];   // (B, 4) f32
    const float* weights = (const float*)d_in[1];   // (8,)   f32
    float* out = (float*)d_out;                     // (B, 4) f32
    float* U   = (float*)d_ws;                      // 16x16 f32 scratch

    const int B     = in_sizes[0] / N_WIRES;
    const int tiles = B / 16;                       // B = 1,000,000 -> 62,500

    build_unitary_kernel<<<1, 32, 0, stream>>>(weights, U);

    const int blocks  = 2048;                       // 16,384 persistent waves
    const int threads = 256;                        // 8 waves / workgroup
    qenc_wmma_kernel<<<blocks, threads, 0, stream>>>(inputs, U, out, tiles);
}